// FDLoss_38955353375166
// MI455X (gfx1250) — compile-verified
//
#include <hip/hip_runtime.h>

typedef float v2f __attribute__((ext_vector_type(2)));
typedef float v4f __attribute__((ext_vector_type(4)));
typedef float v8f __attribute__((ext_vector_type(8)));

#define BLOCK 256
#define NBLOCKS 4096

// Kernel 1: bandwidth-bound streaming partial reduction.
// val(i) = (t<0 && s<0) ? (s - 2t)^2 : s^2   where s=source[i], t=target[i]
__global__ __launch_bounds__(BLOCK) void fd_partial_kernel(
    const float* __restrict__ src, const float* __restrict__ tgt,
    float* __restrict__ partial, long long n4, long long n) {
  __shared__ float red[BLOCK];

  const long long stride = (long long)gridDim.x * BLOCK;
  long long gid = (long long)blockIdx.x * BLOCK + threadIdx.x;

  const v4f* __restrict__ s4 = (const v4f*)src;
  const v4f* __restrict__ t4 = (const v4f*)tgt;

  v4f acc = {0.f, 0.f, 0.f, 0.f};
  for (long long i = gid; i < n4; i += stride) {
    // 16B vector loads, non-temporal: 411MB stream, no reuse -> don't pollute L2
    v4f s = __builtin_nontemporal_load(&s4[i]);
    v4f t = __builtin_nontemporal_load(&t4[i]);
#pragma unroll
    for (int c = 0; c < 4; ++c) {
      float sv = s[c], tv = t[c];
      float d = (tv < 0.f && sv < 0.f) ? (sv - 2.f * tv) : sv;
      acc[c] = fmaf(d, d, acc[c]);
    }
  }
  // scalar tail (N % 4), empty for this problem size but kept for safety
  for (long long i = n4 * 4 + gid; i < n; i += stride) {
    float sv = src[i], tv = tgt[i];
    float d = (tv < 0.f && sv < 0.f) ? (sv - 2.f * tv) : sv;
    acc[0] = fmaf(d, d, acc[0]);
  }

  red[threadIdx.x] = (acc[0] + acc[1]) + (acc[2] + acc[3]);
  __syncthreads();

  // Every wave redundantly reduces the 256 block partials to 32 lane values
  // (keeps EXEC all-1s in every wave for the WMMA below, per ISA restriction).
  const int lane = threadIdx.x & 31;
  float v = 0.f;
#pragma unroll
  for (int k = 0; k < BLOCK; k += 32) v += red[lane + k];

  // 32-lane horizontal sum via V_WMMA_F32_16X16X4_F32 ones-trick:
  //   B = ones(4x16) (layout-invariant), A row m = {v_m, 0, v_{m+16}, 0}
  //   => D[m][*] = v_m + v_{m+16}; summing a lane's 8 D regs gives
  //   rows 0..7 on lanes 0-15 and rows 8..15 on lanes 16-31.
  v2f A; A[0] = v;   A[1] = 0.f;
  v2f B; B[0] = 1.f; B[1] = 1.f;
  v8f C = {};
  v8f D = __builtin_amdgcn_wmma_f32_16x16x4_f32(
      /*neg_a=*/false, A, /*neg_b=*/false, B,
      /*c_mod=*/(short)0, C, /*reuse_a=*/false, /*reuse_b=*/false);

  float t = ((D[0] + D[1]) + (D[2] + D[3])) + ((D[4] + D[5]) + (D[6] + D[7]));
  t += __shfl_xor(t, 16, 32);  // combine the two half-sums -> full 32-lane sum

  if (threadIdx.x == 0) partial[blockIdx.x] = t;
}

// Kernel 2: tiny final reduction in fp64, writes mean.
__global__ __launch_bounds__(BLOCK) void fd_final_kernel(
    const float* __restrict__ partial, int nparts,
    float* __restrict__ out, long long n) {
  __shared__ double red[BLOCK];
  double acc = 0.0;
  for (int i = threadIdx.x; i < nparts; i += BLOCK) acc += (double)partial[i];
  red[threadIdx.x] = acc;
  __syncthreads();
  if (threadIdx.x == 0) {
    double s = 0.0;
#pragma unroll 8
    for (int i = 0; i < BLOCK; ++i) s += red[i];
    out[0] = (float)(s / (double)n);
  }
}

extern "C" void kernel_launch(void* const* d_in, const int* in_sizes, int n_in,
                              void* d_out, int out_size, void* d_ws, size_t ws_size,
                              hipStream_t stream) {
  const float* src = (const float*)d_in[0];  // source
  const float* tgt = (const float*)d_in[1];  // target
  float* out = (float*)d_out;
  float* partial = (float*)d_ws;

  const long long n = (long long)in_sizes[0];
  const long long n4 = n / 4;

  int nblocks = NBLOCKS;
  long long maxb = (long long)(ws_size / sizeof(float));
  if (maxb > 0 && maxb < nblocks) nblocks = (int)maxb;
  if (nblocks < 1) nblocks = 1;

  fd_partial_kernel<<<nblocks, BLOCK, 0, stream>>>(src, tgt, partial, n4, n);
  fd_final_kernel<<<1, BLOCK, 0, stream>>>(partial, nblocks, out, n);
}